// DenseRealSpaceLongRangeKernel_66881230733957
// MI455X (gfx1250) — compile-verified
//
#include <hip/hip_runtime.h>

typedef __attribute__((ext_vector_type(2))) float v2f;
typedef __attribute__((ext_vector_type(8))) float v8f;

#define SPLIT 8

// ---------------------------------------------------------------------------
// Kernel 1: graph ranges. rng[g] = first index i with batch[i] >= g, g=0..B.
// ---------------------------------------------------------------------------
__global__ void k_ranges(const int* __restrict__ batch, int n, int nb,
                         int* __restrict__ rng) {
    int g = threadIdx.x;
    if (g <= nb) {
        int lo = 0, hi = n;
        while (lo < hi) {
            int mid = (lo + hi) >> 1;
            if (batch[mid] < g) lo = mid + 1; else hi = mid;
        }
        rng[g] = lo;
    }
}

// ---------------------------------------------------------------------------
// Per-wave context for the tile loop.
// ---------------------------------------------------------------------------
struct Ctx {
    float pax, pay, paz;
    float c00, c01, c02, c10, c11, c12, c20, c21, c22;   // cell (row-major)
    float i00, i01, i02, i10, i11, i12, i20, i21, i22;   // inverse cell
    float negscr, soft2;
    int   rowLo, rowHi;          // same-graph range of this lane's row
    int   lane, half;
    int   myLo, myHi, n;
    const float* pos;
    const float* source;
};

// Pair energy WITH the diagonal included (self term subtracted in k_final:
// for b==a, d=0 -> f=0 -> s=0 -> w=0 -> r2=soft2 exactly, so the self term
// is the closed-form source_b * exp(-scr*soft)/soft).
template <bool MASKED>
__device__ __forceinline__ float pairE(const Ctx& x, float4 t, int bg) {
    float dx = t.x - x.pax, dy = t.y - x.pay, dz = t.z - x.paz;
    // frac = d @ inv   (f_k = sum_j d_j * inv[j][k])
    const float fx = dx * x.i00 + dy * x.i10 + dz * x.i20;
    const float fy = dx * x.i01 + dy * x.i11 + dz * x.i21;
    const float fz = dx * x.i02 + dy * x.i12 + dz * x.i22;
    const float sx = rintf(fx), sy = rintf(fy), sz = rintf(fz);  // RNE == jnp.round
    // d_wrapped = d - s @ C
    const float wx = dx - (sx * x.c00 + sy * x.c10 + sz * x.c20);
    const float wy = dy - (sx * x.c01 + sy * x.c11 + sz * x.c21);
    const float wz = dz - (sx * x.c02 + sy * x.c12 + sz * x.c22);
    const float r2 = x.soft2 + wx * wx + wy * wy + wz * wz;
    // r2 >= soft2 > 0: raw v_rsq is safe; gives both 1/r and r.
    const float rs = __builtin_amdgcn_rsqf(r2);
    const float r  = r2 * rs;
    const float kv = __expf(x.negscr * r) * rs;
    float e = t.w * kv;
    if (MASKED)    // boundary tiles only: same-graph via sorted-range test
        e = ((bg >= x.rowLo) & (bg < x.rowHi)) ? e : 0.0f;
    return e;
}

template <bool MASKED>
__device__ __forceinline__ v8f tile_loop(const Ctx& x, float4* sT,
                                         v8f acc, v2f onesB) {
    for (int base = x.myLo; base < x.myHi; base += 32) {
        const int bi = base + x.lane;
        const int bc = min(bi, x.n - 1);
        float4 t;
        t.x = x.pos[bc * 3 + 0];
        t.y = x.pos[bc * 3 + 1];
        t.z = x.pos[bc * 3 + 2];
        t.w = (bi < x.myHi) ? x.source[bc] : 0.0f;  // split bound folded in
        __syncthreads();          // protect previous iteration's LDS reads
        sT[x.lane] = t;
        __syncthreads();
#pragma unroll
        for (int cc = 0; cc < 8; ++cc) {
            const int l0 = cc * 4 + x.half * 2;   // A layout: K=(half?2:0), K+1
            const int l1 = l0 + 1;
            v2f A;
            A.x = pairE<MASKED>(x, sT[l0], base + l0);  // VGPR0 -> K0 / K2
            A.y = pairE<MASKED>(x, sT[l1], base + l1);  // VGPR1 -> K1 / K3
            // D = A(16x4) x Ones(4x16) + C : every column of D = row sums
            acc = __builtin_amdgcn_wmma_f32_16x16x4_f32(
                false, A, false, onesB, (short)0, acc, false, false);
        }
    }
    return acc;
}

// ---------------------------------------------------------------------------
// Kernel 2: one wave per (16-row a-tile, b-split). Pair energies are computed
// directly in the V_WMMA_F32_16X16X4_F32 A-matrix layout; B = all-ones so the
// WMMA performs the column-sum reduction into the f32 C accumulator.
// ---------------------------------------------------------------------------
__global__ __launch_bounds__(32)
void k_pairs(const float* __restrict__ pos, const int* __restrict__ batch,
             const float* __restrict__ cell, const float* __restrict__ source,
             const float* __restrict__ scrp, const float* __restrict__ softp,
             const int* __restrict__ rng, float* __restrict__ partial, int n) {
    const int lane = threadIdx.x;
    const int m    = lane & 15;
    const int tileBase = blockIdx.x * 16;
    const int row = tileBase + m;

    Ctx x;
    x.lane = lane;
    x.half = lane >> 4;
    x.n = n;
    x.pos = pos;
    x.source = source;

    x.pax = pos[row * 3 + 0];
    x.pay = pos[row * 3 + 1];
    x.paz = pos[row * 3 + 2];
    const int ga = batch[row];
    x.rowLo = rng[ga];
    x.rowHi = rng[ga + 1];

    const float* C = cell + ga * 9;
    x.c00 = C[0]; x.c01 = C[1]; x.c02 = C[2];
    x.c10 = C[3]; x.c11 = C[4]; x.c12 = C[5];
    x.c20 = C[6]; x.c21 = C[7]; x.c22 = C[8];
    // adjugate inverse
    float i00 = x.c11 * x.c22 - x.c12 * x.c21;
    float i01 = x.c02 * x.c21 - x.c01 * x.c22;
    float i02 = x.c01 * x.c12 - x.c02 * x.c11;
    float i10 = x.c12 * x.c20 - x.c10 * x.c22;
    float i11 = x.c00 * x.c22 - x.c02 * x.c20;
    float i12 = x.c02 * x.c10 - x.c00 * x.c12;
    float i20 = x.c10 * x.c21 - x.c11 * x.c20;
    float i21 = x.c01 * x.c20 - x.c00 * x.c21;
    float i22 = x.c00 * x.c11 - x.c01 * x.c10;
    const float rdet = 1.0f / (x.c00 * i00 + x.c01 * i10 + x.c02 * i20);
    x.i00 = i00 * rdet; x.i01 = i01 * rdet; x.i02 = i02 * rdet;
    x.i10 = i10 * rdet; x.i11 = i11 * rdet; x.i12 = i12 * rdet;
    x.i20 = i20 * rdet; x.i21 = i21 * rdet; x.i22 = i22 * rdet;

    x.negscr = -scrp[0];
    const float soft = softp[0];
    x.soft2 = soft * soft;

    // wave-uniform column range (batch sorted)
    const int gFirst = __builtin_amdgcn_readlane(ga, 0);
    const int gLast  = __builtin_amdgcn_readlane(ga, 15);
    const int bLo = rng[gFirst];
    const int bHi = rng[gLast + 1];
    const int total = bHi - bLo;
    const int per   = ((total + SPLIT * 32 - 1) / (SPLIT * 32)) * 32;
    x.myLo = bLo + (int)blockIdx.y * per;
    x.myHi = min(bHi, x.myLo + per);

    __shared__ float4 sT[32];

    v8f acc = {};
    v2f onesB; onesB.x = 1.0f; onesB.y = 1.0f;

    // Uniform-graph tiles (the common case): mask is provably always true,
    // compile it out. Boundary tiles take the masked path.
    if (gFirst == gLast)
        acc = tile_loop<false>(x, sT, acc, onesB);
    else
        acc = tile_loop<true>(x, sT, acc, onesB);

    // C/D layout: lane 0 holds M=0..7 in VGPR 0..7, lane 16 holds M=8..15.
    float* myPart = partial + (size_t)blockIdx.y * n + tileBase;
    if (lane == 0) {
#pragma unroll
        for (int r = 0; r < 8; ++r) myPart[r] = acc[r];
    }
    if (lane == 16) {
#pragma unroll
        for (int r = 0; r < 8; ++r) myPart[8 + r] = acc[r];
    }
}

// ---------------------------------------------------------------------------
// Kernel 3: fixed-order reduction of SPLIT partials + self-term subtraction.
// ---------------------------------------------------------------------------
__global__ void k_final(const float* __restrict__ partial,
                        const float* __restrict__ source,
                        const float* __restrict__ scrp,
                        const float* __restrict__ softp,
                        float* __restrict__ out, int n) {
    const int i = blockIdx.x * blockDim.x + threadIdx.x;
    if (i < n) {
        const float soft  = softp[0];
        const float soft2 = soft * soft;
        // identical arithmetic to the loop's self pair (r2 == soft2 exactly)
        const float rs = __builtin_amdgcn_rsqf(soft2);
        const float r  = soft2 * rs;
        const float selfK = __expf(-scrp[0] * r) * rs;
        float s = 0.0f;
        for (int y = 0; y < SPLIT; ++y) s += partial[(size_t)y * n + i];
        const float src = source[i];
        out[i] = 0.5f * src * (s - src * selfK);
    }
}

extern "C" void kernel_launch(void* const* d_in, const int* in_sizes, int n_in,
                              void* d_out, int out_size, void* d_ws, size_t ws_size,
                              hipStream_t stream) {
    const float* pos    = (const float*)d_in[0];
    const int*   batch  = (const int*)d_in[1];
    const float* cell   = (const float*)d_in[2];
    const float* source = (const float*)d_in[3];
    const float* scr    = (const float*)d_in[4];
    const float* soft   = (const float*)d_in[5];

    const int n  = in_sizes[0] / 3;   // N atoms
    const int nb = in_sizes[2] / 9;   // B graphs

    int*   rng     = (int*)d_ws;                         // nb+1 ints
    float* partial = (float*)((char*)d_ws + 256);        // SPLIT * n floats

    k_ranges<<<1, 32, 0, stream>>>(batch, n, nb, rng);
    dim3 grid(n / 16, SPLIT);
    k_pairs<<<grid, 32, 0, stream>>>(pos, batch, cell, source, scr, soft,
                                     rng, partial, n);
    k_final<<<(n + 255) / 256, 256, 0, stream>>>(partial, source, scr, soft,
                                                 (float*)d_out, n);
}